// SoundWave2DModel_58351425683794
// MI455X (gfx1250) — compile-verified
//
#include <hip/hip_runtime.h>
#include <cstdint>

// Problem constants (from reference)
#define NZ   2048
#define NX   2048
#define NT   32
#define NCH  128
#define DT2  0.09f            // DT*DT, DT = 0.3, DZ = 1.0

// Tiling: tile = 64 (x) x 16 (z); LDS row = [x0-4, x0+68) so every 4-float
// chunk is 16B-aligned and fully in- or out-of-range (x0 multiple of 64).
#define TX     64
#define TZ     16
#define LDSW   72             // 4 left halo + 64 + 4 right halo (stencil uses +-2)
#define LDSH   20             // 2 + 16 + 2
#define NCHUNK (LDSW / 4)     // 18 b128 chunks per row
#define NQ     (LDSH * NCHUNK) // 360 chunks per tile

typedef int v4i __attribute__((__vector_size__(16)));
typedef __attribute__((address_space(1))) v4i as1_v4i;   // "__device__ v4i*"
typedef __attribute__((address_space(3))) v4i as3_v4i;   // "__shared__ v4i*"

// --- CDNA5 async global->LDS DMA (ASYNCcnt path). Builtin if present,
// --- otherwise raw gfx1250 mnemonic per CDNA5 ISA.
__device__ __forceinline__ void async_ld_b128(const float* g, float* l) {
#if __has_builtin(__builtin_amdgcn_global_load_async_to_lds_b128)
  __builtin_amdgcn_global_load_async_to_lds_b128((as1_v4i*)g, (as3_v4i*)l,
                                                 /*imm offset*/0, /*cpol*/0);
#else
  unsigned lds_off = (unsigned)(uintptr_t)(as3_v4i*)l;
  unsigned long long ga = (unsigned long long)(uintptr_t)g;
  asm volatile("global_load_async_to_lds_b128 %0, %1, off"
               :: "v"(lds_off), "v"(ga) : "memory");
#endif
}

__device__ __forceinline__ void wait_async_zero() {
#if __has_builtin(__builtin_amdgcn_s_wait_asynccnt)
  __builtin_amdgcn_s_wait_asynccnt(0);
#else
  asm volatile("s_wait_asynccnt 0" ::: "memory");
#endif
}

// ---------------------------------------------------------------------------
// Precompute time-invariant fields (one pass over the grid), zero p buffers.
//   dF  = 2*beta / (v^2*rho)
//   CL  = dt2 * K * rho                  (multiplies raw Laplacian, DZ=1)
//   GXQ = dt2 * K * gx_ir * 0.5          (multiplies pm1[x+1]-pm1[x-1])
//   GZQ = dt2 * K * gz_ir * 0.5          (multiplies pm1[z+1]-pm1[z-1])
// gx_ir/gz_ir are central differences of 1/rho with zero padding (as in ref).
// ---------------------------------------------------------------------------
__global__ __launch_bounds__(256) void precompute_kernel(
    const float* __restrict__ vel, const float* __restrict__ den,
    const float* __restrict__ beta,
    float* __restrict__ dF, float* __restrict__ CL,
    float* __restrict__ GXQ, float* __restrict__ GZQ,
    float* __restrict__ pA, float* __restrict__ pB)
{
  const long long idx = (long long)blockIdx.x * blockDim.x + threadIdx.x;
  if (idx >= (long long)NZ * NX) return;
  const int x = (int)(idx & (NX - 1));
  const int z = (int)(idx >> 11);          // NX = 2048 = 2^11

  const float v   = vel[idx];
  const float rho = den[idx];
  const float bet = beta[idx];
  const float K   = v * v * rho;

  const float invL = (x > 0)      ? 1.0f / den[idx - 1]  : 0.0f;
  const float invR = (x < NX - 1) ? 1.0f / den[idx + 1]  : 0.0f;
  const float invU = (z > 0)      ? 1.0f / den[idx - NX] : 0.0f;
  const float invD = (z < NZ - 1) ? 1.0f / den[idx + NX] : 0.0f;
  const float gx_ir = 0.5f * (invR - invL);   // conv(1/rho, GX)
  const float gz_ir = 0.5f * (invD - invU);   // conv(1/rho, GZ)

  const float dt2K = DT2 * K;
  dF[idx]  = 2.0f * bet / K;
  CL[idx]  = dt2K * rho;
  GXQ[idx] = dt2K * gx_ir * 0.5f;
  GZQ[idx] = dt2K * gz_ir * 0.5f;
  pA[idx]  = 0.0f;
  pB[idx]  = 0.0f;
}

// ---------------------------------------------------------------------------
// One FDTD step. pm1 stencil tile staged into LDS with async b128 DMA,
// pointwise fields prefetched into registers concurrently with the DMA.
// Writes p in-place over pm2 buffer (pointwise-safe).
// Block = (64,4) = 256 threads (8 wave32s); tile = 64 (x) x 16 (z).
// ---------------------------------------------------------------------------
__global__ __launch_bounds__(256) void step_kernel(
    const float* __restrict__ pm1, float* __restrict__ pm2io,
    const float* __restrict__ dF, const float* __restrict__ damping,
    const float* __restrict__ CL, const float* __restrict__ GXQ,
    const float* __restrict__ GZQ,
    const float* __restrict__ pulse, int t)
{
  __shared__ float tile[LDSH * LDSW];      // 5.76 KB

  const int x0  = blockIdx.x * TX;
  const int z0  = blockIdx.y * TZ;
  const int tid = threadIdx.y * 64 + threadIdx.x;

  // ---- Stage pm1 halo tile (20 rows x 72 cols) via async b128 DMA ----------
  // Chunk q -> row lr = q/18, 4-float chunk lc4 = q%18.
  // Global span of chunk: x in [x0-4+4*lc4, x0-4+4*lc4+3], z = z0-2+lr.
  // Every chunk is fully in-range or fully out-of-range (zero padding).
  for (int q = tid; q < NQ; q += 256) {
    const int lr  = q / NCHUNK;
    const int lc4 = q - lr * NCHUNK;
    const int gz  = z0 + lr - 2;
    const int gxs = x0 + 4 * lc4 - 4;
    float* lp = &tile[lr * LDSW + 4 * lc4];
    if (gz >= 0 && gz < NZ && gxs >= 0 && gxs + 3 < NX) {
      async_ld_b128(pm1 + ((long long)gz * NX + gxs), lp);
    } else {
      *(float4*)lp = make_float4(0.f, 0.f, 0.f, 0.f);   // ds_store_b128 zeros
    }
  }

  // ---- Overlap: issue pointwise-field loads while the DMA is in flight -----
  const float pt = DT2 * pulse[t];         // uniform source term (s_load)
  float pm2v[4], dv[4], damp[4], clv[4], gxv[4], gzv[4];
  long long gidx[4];
  #pragma unroll
  for (int r = 0; r < 4; ++r) {
    const int lz = threadIdx.y * 4 + r;
    const long long g = (long long)(z0 + lz) * NX + (x0 + threadIdx.x);
    gidx[r] = g;
    pm2v[r] = pm2io[g];
    dv[r]   = dF[g];
    damp[r] = damping[g];
    clv[r]  = CL[g];
    gxv[r]  = GXQ[g];
    gzv[r]  = GZQ[g];
  }

  wait_async_zero();                       // drain ASYNCcnt before barrier
  __syncthreads();

  // 4th-order cross-Laplacian coefficients (raw; dt2*K*rho folded into CL)
  const float C0 = -5.0f;
  const float C1 = 4.0f / 3.0f;
  const float C2 = -1.0f / 12.0f;

  #pragma unroll
  for (int r = 0; r < 4; ++r) {
    const int lz = threadIdx.y * 4 + r;
    const int c  = (lz + 2) * LDSW + (threadIdx.x + 4);   // tile center

    const float pc  = tile[c];
    const float lap = C0 * pc
        + C1 * (tile[c - 1] + tile[c + 1] + tile[c - LDSW] + tile[c + LDSW])
        + C2 * (tile[c - 2] + tile[c + 2] + tile[c - 2 * LDSW] + tile[c + 2 * LDSW]);
    const float gxd = tile[c + 1]    - tile[c - 1];       // *GXQ (0.5 folded)
    const float gzd = tile[c + LDSW] - tile[c - LDSW];    // *GZQ (0.5 folded)

    const float nla  = 1.0f + dv[r] * pc;
    const float a    = nla + 2.0f * damp[r];
    const float b    = 2.0f * nla - damp[r] * damp[r];
    const float cc   = 2.0f * damp[r] - nla;
    const float diff = pc - pm2v[r];   // d*(2*p1*p2 - p1^2 - p2^2) = -d*diff^2

    const float num = b * pc + cc * pm2v[r] - dv[r] * diff * diff
                    + clv[r] * lap + gxv[r] * gxd + gzv[r] * gzd + pt;
    pm2io[gidx[r]] = num / a;
  }
}

// ---------------------------------------------------------------------------
// Gather 128 receiver samples for timestep t.
// ---------------------------------------------------------------------------
__global__ void gather_kernel(const float* __restrict__ p,
                              const int* __restrict__ loc,
                              float* __restrict__ out, int t)
{
  const int ch = threadIdx.x;
  if (ch < NCH) {
    const int rz = loc[2 * ch + 0];
    const int cx = loc[2 * ch + 1];
    out[t * NCH + ch] = p[(long long)rz * NX + cx];
  }
}

// ---------------------------------------------------------------------------
extern "C" void kernel_launch(void* const* d_in, const int* in_sizes, int n_in,
                              void* d_out, int out_size, void* d_ws, size_t ws_size,
                              hipStream_t stream)
{
  const float* vel     = (const float*)d_in[0];
  const float* den     = (const float*)d_in[1];
  const float* damping = (const float*)d_in[2];
  const float* beta    = (const float*)d_in[3];
  const float* pulse   = (const float*)d_in[4];
  const int*   loc     = (const int*)  d_in[5];
  float* out = (float*)d_out;

  const long long N = (long long)NZ * NX;
  float* ws  = (float*)d_ws;        // 6*N floats = 96 MB of scratch
  float* dF  = ws + 0 * N;
  float* CLf = ws + 1 * N;
  float* GXQ = ws + 2 * N;
  float* GZQ = ws + 3 * N;
  float* pA  = ws + 4 * N;
  float* pB  = ws + 5 * N;

  {
    const int threads = 256;
    const int blocks  = (int)((N + threads - 1) / threads);
    precompute_kernel<<<blocks, threads, 0, stream>>>(
        vel, den, beta, dF, CLf, GXQ, GZQ, pA, pB);
  }

  dim3 grid(NX / TX, NZ / TZ);      // 32 x 128 blocks
  dim3 block(64, 4);                // 256 threads = 8 wave32s

  float* pm1 = pA;                  // both zero-initialized
  float* pm2 = pB;
  for (int t = 0; t < NT; ++t) {
    step_kernel<<<grid, block, 0, stream>>>(
        pm1, pm2, dF, damping, CLf, GXQ, GZQ, pulse, t);
    gather_kernel<<<1, NCH, 0, stream>>>(pm2, loc, out, t);
    float* tmp = pm1; pm1 = pm2; pm2 = tmp;   // new carry = (p, old pm1)
  }
}